// DGCNN_53506702574037
// MI455X (gfx1250) — compile-verified
//
#include <hip/hip_runtime.h>
#include <math.h>

#define KNN_K 20
#define EPS 1e-05f
#define SLOPE 0.2f
#define NPTS 1024
#define NB 8

typedef __attribute__((ext_vector_type(2))) float v2f;
typedef __attribute__((ext_vector_type(8))) float v8f;

__device__ __forceinline__ float lrelu(float y) { return y > 0.f ? y : SLOPE * y; }

__device__ __forceinline__ v8f wmma4(v2f a, v2f b, v8f c) {
    return __builtin_amdgcn_wmma_f32_16x16x4_f32(false, a, false, b, (short)0, c,
                                                 false, false);
}

// ---------------------------------------------------------------------------
// xx[b,n] = sum_c x[b,c,n]^2
// ---------------------------------------------------------------------------
__global__ void xx_kernel(const float* __restrict__ X, size_t bstride, int C,
                          float* __restrict__ xx) {
    constexpr int N = NPTS;
    int i = blockIdx.x * blockDim.x + threadIdx.x;
    if (i >= NB * N) return;
    int b = i / N, n = i % N;
    const float* xb = X + (size_t)b * bstride + n;
    float s = 0.f;
    for (int c = 0; c < C; ++c) { float v = xb[(size_t)c * N]; s += v * v; }
    xx[i] = s;
}

// ---------------------------------------------------------------------------
// Pairwise "neg. squared distance": D[b,n,m] = 2*G[n,m] - xx[n] - xx[m].
// G = X^T X via V_WMMA_F32_16X16X4_F32; 16(n) x 64(m) per wave32.
// N compile-time (1024) so all row offsets fold into load immediates;
// unsigned 32-bit offsets from a uniform base keep address math scalar+u32.
// ---------------------------------------------------------------------------
__global__ void __launch_bounds__(32)
gram_dist_kernel(const float* __restrict__ X, size_t bstride, int C,
                 const float* __restrict__ xx, float* __restrict__ D) {
    constexpr int N = NPTS;
    int b = blockIdx.z;
    int n0 = blockIdx.y * 16;          // row tile
    int m0 = blockIdx.x * 64;          // 4 col tiles
    int lane = threadIdx.x;
    int half = lane >> 4;
    int l = lane & 15;
    const float* Xb = X + (size_t)b * bstride;
    unsigned aoff = (unsigned)(2 * half) * N + n0 + l;   // A rows: [0], [N]
    unsigned boff = (unsigned)(2 * half) * N + m0 + l;   // B rows: [0], [N]
    v8f acc0 = {0.f,0.f,0.f,0.f,0.f,0.f,0.f,0.f};
    v8f acc1 = acc0, acc2 = acc0, acc3 = acc0;
    int c0 = 0;
    for (; c0 + 4 <= C; c0 += 4) {                 // unguarded hot loop
        v2f a  = {Xb[aoff],      Xb[aoff + N]};
        v2f b0 = {Xb[boff],      Xb[boff + N]};
        v2f b1 = {Xb[boff + 16], Xb[boff + N + 16]};
        v2f b2 = {Xb[boff + 32], Xb[boff + N + 32]};
        v2f b3 = {Xb[boff + 48], Xb[boff + N + 48]};
        acc0 = wmma4(a, b0, acc0);
        acc1 = wmma4(a, b1, acc1);
        acc2 = wmma4(a, b2, acc2);
        acc3 = wmma4(a, b3, acc3);
        aoff += 4 * N; boff += 4 * N;
    }
    if (c0 < C) {                                   // tail: only C=3 hits this
        int ca = c0 + 2 * half;
        bool v0 = ca < C, v1 = ca + 1 < C;
        v2f a, b0, b1, b2, b3;
        a[0]  = v0 ? Xb[aoff]      : 0.f;  a[1]  = v1 ? Xb[aoff + N]      : 0.f;
        b0[0] = v0 ? Xb[boff]      : 0.f;  b0[1] = v1 ? Xb[boff + N]      : 0.f;
        b1[0] = v0 ? Xb[boff + 16] : 0.f;  b1[1] = v1 ? Xb[boff + N + 16] : 0.f;
        b2[0] = v0 ? Xb[boff + 32] : 0.f;  b2[1] = v1 ? Xb[boff + N + 32] : 0.f;
        b3[0] = v0 ? Xb[boff + 48] : 0.f;  b3[1] = v1 ? Xb[boff + N + 48] : 0.f;
        acc0 = wmma4(a, b0, acc0);
        acc1 = wmma4(a, b1, acc1);
        acc2 = wmma4(a, b2, acc2);
        acc3 = wmma4(a, b3, acc3);
    }
    const float* xxb = xx + (size_t)b * N;
    float xm0 = xxb[m0 + l], xm1 = xxb[m0 + l + 16];
    float xm2 = xxb[m0 + l + 32], xm3 = xxb[m0 + l + 48];
    float* Db = D + (size_t)b * N * N;
    unsigned doff = (unsigned)(n0 + 8 * half) * N + m0 + l;
#pragma unroll
    for (int r = 0; r < 8; ++r) {
        float xr2 = xxb[n0 + r + 8 * half];
        Db[doff]      = 2.f * acc0[r] - xr2 - xm0;
        Db[doff + 16] = 2.f * acc1[r] - xr2 - xm1;
        Db[doff + 32] = 2.f * acc2[r] - xr2 - xm2;
        Db[doff + 48] = 2.f * acc3[r] - xr2 - xm3;
        doff += N;
    }
}

// ---------------------------------------------------------------------------
// Top-K (K=20, largest, lowest-index tie-break) per length-N row.
// 8 waves/block, one row per wave, rows staged in LDS (32 KB of 320 KB/WGP).
// ---------------------------------------------------------------------------
__global__ void __launch_bounds__(256)
topk_kernel(const float* __restrict__ D, int* __restrict__ idxout) {
    constexpr int N = NPTS;
    __shared__ float sv[8][NPTS];
    int wave = threadIdx.x >> 5;
    int lane = threadIdx.x & 31;
    int row = blockIdx.x * 8 + wave;   // b*N + n
    const float* d = D + (size_t)row * N;
    float* s = sv[wave];
    for (int j = lane; j < N; j += 32) s[j] = d[j];
    __syncthreads();
    for (int kk = 0; kk < KNN_K; ++kk) {
        float bv = -INFINITY;
        int bi = N;
        for (int t = 0; t < N / 32; ++t) {
            int j = t * 32 + lane;
            float x = s[j];
            if (x > bv || (x == bv && j < bi)) { bv = x; bi = j; }
        }
#pragma unroll
        for (int off = 16; off > 0; off >>= 1) {
            float ov = __shfl_xor(bv, off, 32);
            int   oi = __shfl_xor(bi, off, 32);
            if (ov > bv || (ov == bv && oi < bi)) { bv = ov; bi = oi; }
        }
        if (lane == 0) idxout[(size_t)row * KNN_K + kk] = bi;
        s[bi] = -INFINITY;             // all lanes store same value: benign
        __syncthreads();
    }
}

// ---------------------------------------------------------------------------
// wc[o,c] = w[o, C+c] - w[o, c]
// ---------------------------------------------------------------------------
__global__ void wcomb_kernel(const float* __restrict__ w, float* __restrict__ wc,
                             int O, int C) {
    int i = blockIdx.x * blockDim.x + threadIdx.x;
    if (i >= O * C) return;
    int o = i / C, c = i % C;
    wc[i] = w[(size_t)o * 2 * C + C + c] - w[(size_t)o * 2 * C + c];
}

// ---------------------------------------------------------------------------
// Y[b,o,n] = sum_c W[o,c] * X[b,c,n].  16(o) x 64(n) per wave32:
// one A (weight) operand reused across 4 B tiles, 4 accumulators.
// ---------------------------------------------------------------------------
__global__ void __launch_bounds__(32)
wmma_gemm_kernel(const float* __restrict__ W, int ldw,
                 const float* __restrict__ X, size_t bstride,
                 float* __restrict__ Y, int O, int C) {
    constexpr int N = NPTS;
    int b = blockIdx.z;
    int n0 = blockIdx.x * 64;
    int o0 = blockIdx.y * 16;
    int lane = threadIdx.x;
    int half = lane >> 4;
    int l = lane & 15;
    const float* Xb = X + (size_t)b * bstride;
    unsigned woff = (unsigned)(o0 + l) * ldw + 2 * half;  // A: [0], [1]
    unsigned xoff = (unsigned)(2 * half) * N + n0 + l;    // B rows: [0], [N]
    v8f acc0 = {0.f,0.f,0.f,0.f,0.f,0.f,0.f,0.f};
    v8f acc1 = acc0, acc2 = acc0, acc3 = acc0;
    int c0 = 0;
    for (; c0 + 4 <= C; c0 += 4) {                 // unguarded hot loop
        float2 aw = *reinterpret_cast<const float2*>(W + woff); // 8B-aligned here
        v2f a  = {aw.x, aw.y};
        v2f b0 = {Xb[xoff],      Xb[xoff + N]};
        v2f b1 = {Xb[xoff + 16], Xb[xoff + N + 16]};
        v2f b2 = {Xb[xoff + 32], Xb[xoff + N + 32]};
        v2f b3 = {Xb[xoff + 48], Xb[xoff + N + 48]};
        acc0 = wmma4(a, b0, acc0);
        acc1 = wmma4(a, b1, acc1);
        acc2 = wmma4(a, b2, acc2);
        acc3 = wmma4(a, b3, acc3);
        woff += 4; xoff += 4 * N;
    }
    if (c0 < C) {                                   // tail: only C=3 hits this
        int ca = c0 + 2 * half;
        bool v0 = ca < C, v1 = ca + 1 < C;
        v2f a, b0, b1, b2, b3;
        a[0]  = v0 ? W[woff]        : 0.f;  a[1]  = v1 ? W[woff + 1]       : 0.f;
        b0[0] = v0 ? Xb[xoff]       : 0.f;  b0[1] = v1 ? Xb[xoff + N]      : 0.f;
        b1[0] = v0 ? Xb[xoff + 16]  : 0.f;  b1[1] = v1 ? Xb[xoff + N + 16] : 0.f;
        b2[0] = v0 ? Xb[xoff + 32]  : 0.f;  b2[1] = v1 ? Xb[xoff + N + 32] : 0.f;
        b3[0] = v0 ? Xb[xoff + 48]  : 0.f;  b3[1] = v1 ? Xb[xoff + N + 48] : 0.f;
        acc0 = wmma4(a, b0, acc0);
        acc1 = wmma4(a, b1, acc1);
        acc2 = wmma4(a, b2, acc2);
        acc3 = wmma4(a, b3, acc3);
    }
    float* Yb = Y + (size_t)b * O * N;
    unsigned yoff = (unsigned)(o0 + 8 * half) * N + n0 + l;
#pragma unroll
    for (int r = 0; r < 8; ++r) {
        Yb[yoff]      = acc0[r];
        Yb[yoff + 16] = acc1[r];
        Yb[yoff + 32] = acc2[r];
        Yb[yoff + 48] = acc3[r];
        yoff += N;
    }
}

// ---------------------------------------------------------------------------
// out[b, catOff+o, n] = lrelu(bn( max_k z[b,o,idx[b,n,k]] + t[b,o,n] ))
// ---------------------------------------------------------------------------
__global__ void edge_max_bn_kernel(const float* __restrict__ z, const float* __restrict__ t,
                                   const int* __restrict__ idx, const float* __restrict__ bnp,
                                   float* __restrict__ out, int O,
                                   int catC, int catOff) {
    constexpr int N = NPTS;
    int i = blockIdx.x * blockDim.x + threadIdx.x;
    if (i >= NB * O * N) return;
    int n = i % N;
    int rest = i / N;
    int o = rest % O;
    int b = rest / O;
    const float* zr = z + ((size_t)b * O + o) * N;
    const int* ir = idx + ((size_t)b * N + n) * KNN_K;
    float m = -INFINITY;
#pragma unroll
    for (int k = 0; k < KNN_K; ++k) m = fmaxf(m, zr[ir[k]]);
    m += t[((size_t)b * O + o) * N + n];
    float g = bnp[o], bb = bnp[O + o], mu = bnp[2 * O + o], var = bnp[3 * O + o];
    float y = (m - mu) * (g * rsqrtf(var + EPS)) + bb;
    out[((size_t)b * catC + catOff + o) * N + n] = lrelu(y);
}

// ---------------------------------------------------------------------------
// Pooling over n with fused BN+LReLU: h[b,o]=max, h[b,O+o]=mean. One wave/(b,o).
// ---------------------------------------------------------------------------
__global__ void __launch_bounds__(256)
pool_kernel(const float* __restrict__ e, const float* __restrict__ bnp,
            float* __restrict__ h, int O) {
    constexpr int N = NPTS;
    int wave = threadIdx.x >> 5;
    int lane = threadIdx.x & 31;
    int bo = blockIdx.x * 8 + wave;
    int b = bo / O, o = bo % O;
    const float* row = e + ((size_t)b * O + o) * N;
    float s = bnp[o] * rsqrtf(bnp[3 * O + o] + EPS);
    float mu = bnp[2 * O + o], bb = bnp[O + o];
    float mx = -INFINITY, sm = 0.f;
    for (int j = lane; j < N; j += 32) {
        float y = lrelu((row[j] - mu) * s + bb);
        mx = fmaxf(mx, y);
        sm += y;
    }
#pragma unroll
    for (int off = 16; off > 0; off >>= 1) {
        mx = fmaxf(mx, __shfl_xor(mx, off, 32));
        sm += __shfl_xor(sm, off, 32);
    }
    if (lane == 0) {
        h[(size_t)b * 2 * O + o] = mx;
        h[(size_t)b * 2 * O + O + o] = sm / (float)N;
    }
}

// ---------------------------------------------------------------------------
// Small FC head
// ---------------------------------------------------------------------------
__global__ void fc_kernel(const float* __restrict__ in, const float* __restrict__ W,
                          const float* __restrict__ bnp, const float* __restrict__ bias,
                          float* __restrict__ out, int B, int Cin, int O, int mode) {
    int i = blockIdx.x * blockDim.x + threadIdx.x;
    if (i >= B * O) return;
    int b = i / O, o = i % O;
    const float* x = in + (size_t)b * Cin;
    const float* wr = W + (size_t)o * Cin;
    float acc = 0.f;
    for (int c = 0; c < Cin; ++c) acc += x[c] * wr[c];
    if (mode == 0) {
        float g = bnp[o], bb = bnp[O + o], mu = bnp[2 * O + o], var = bnp[3 * O + o];
        acc = lrelu((acc - mu) * (g * rsqrtf(var + EPS)) + bb);
    } else {
        acc += bias[o];
    }
    out[(size_t)b * O + o] = acc;
}

// ---------------------------------------------------------------------------
extern "C" void kernel_launch(void* const* d_in, const int* in_sizes, int n_in,
                              void* d_out, int out_size, void* d_ws, size_t ws_size,
                              hipStream_t stream) {
    (void)in_sizes; (void)n_in; (void)out_size; (void)ws_size;
    const int N = NPTS, B = NB;
    const float* x    = (const float*)d_in[0];
    const float* w0   = (const float*)d_in[1];
    const float* w1   = (const float*)d_in[2];
    const float* w2   = (const float*)d_in[3];
    const float* w3   = (const float*)d_in[4];
    const float* bn0  = (const float*)d_in[5];
    const float* bn1  = (const float*)d_in[6];
    const float* bn2  = (const float*)d_in[7];
    const float* bn3  = (const float*)d_in[8];
    const float* we   = (const float*)d_in[9];
    const float* bne  = (const float*)d_in[10];
    const float* wf0  = (const float*)d_in[11];
    const float* bnf0 = (const float*)d_in[12];
    const float* wf1  = (const float*)d_in[13];
    const float* bnf1 = (const float*)d_in[14];
    const float* wfin = (const float*)d_in[15];
    const float* bfin = (const float*)d_in[16];
    float* out = (float*)d_out;
    float* ws = (float*)d_ws;

    // workspace layout (floats); total ~17.0M floats (~68 MB), L2-resident
    size_t o_cat  = 0;                              // [B,512,N]
    size_t o_dist = o_cat  + (size_t)B * 512 * N;   // [B,N,N] (reused for e)
    size_t o_xx   = o_dist + (size_t)B * N * N;     // [B,N]
    size_t o_z    = o_xx   + (size_t)B * N;         // [B,256,N]
    size_t o_t    = o_z    + (size_t)B * 256 * N;   // [B,256,N]
    size_t o_wc   = o_t    + (size_t)B * 256 * N;   // [256*256]
    size_t o_h    = o_wc   + 65536;                 // [B,2048]
    size_t o_h1   = o_h    + (size_t)B * 2048;      // [B,512]
    size_t o_h2   = o_h1   + (size_t)B * 512;       // [B,256]
    size_t o_idx  = o_h2   + (size_t)B * 256;       // [B,N,K] ints

    float* cat  = ws + o_cat;
    float* dist = ws + o_dist;
    float* xxb  = ws + o_xx;
    float* zb   = ws + o_z;
    float* tb   = ws + o_t;
    float* wcb  = ws + o_wc;
    float* hb   = ws + o_h;
    float* h1b  = ws + o_h1;
    float* h2b  = ws + o_h2;
    int*   idxb = (int*)(ws + o_idx);

    auto conv_block = [&](const float* Xin, size_t bstride, int Cin, int O,
                          const float* w, const float* bnp, int catOff) {
        xx_kernel<<<(B * N + 255) / 256, 256, 0, stream>>>(Xin, bstride, Cin, xxb);
        dim3 gG(N / 64, N / 16, B);
        gram_dist_kernel<<<gG, 32, 0, stream>>>(Xin, bstride, Cin, xxb, dist);
        topk_kernel<<<B * N / 8, 256, 0, stream>>>(dist, idxb);
        wcomb_kernel<<<(O * Cin + 255) / 256, 256, 0, stream>>>(w, wcb, O, Cin);
        dim3 gM(N / 64, O / 16, B);
        wmma_gemm_kernel<<<gM, 32, 0, stream>>>(w,   2 * Cin, Xin, bstride, zb, O, Cin);
        wmma_gemm_kernel<<<gM, 32, 0, stream>>>(wcb, Cin,     Xin, bstride, tb, O, Cin);
        edge_max_bn_kernel<<<(B * O * N + 255) / 256, 256, 0, stream>>>(
            zb, tb, idxb, bnp, cat, O, 512, catOff);
    };

    // 4 EdgeConv blocks; outputs land as contiguous slices of `cat` [B,512,N]
    conv_block(x,                     (size_t)3 * N,   3,   64,  w0, bn0, 0);
    conv_block(cat,                   (size_t)512 * N, 64,  64,  w1, bn1, 64);
    conv_block(cat + (size_t)64  * N, (size_t)512 * N, 64,  128, w2, bn2, 128);
    conv_block(cat + (size_t)128 * N, (size_t)512 * N, 128, 256, w3, bn3, 256);

    // embedding GEMM: e[b,o,n] = we[o,:]·cat[b,:,n], O=1024, C=512 (reuse dist buf)
    dim3 gE(N / 64, 1024 / 16, B);
    wmma_gemm_kernel<<<gE, 32, 0, stream>>>(we, 512, cat, (size_t)512 * N, dist, 1024, 512);
    pool_kernel<<<B * 1024 / 8, 256, 0, stream>>>(dist, bne, hb, 1024);

    // FC head
    fc_kernel<<<(B * 512 + 255) / 256, 256, 0, stream>>>(hb,  wf0,  bnf0, nullptr, h1b, B, 2048, 512, 0);
    fc_kernel<<<(B * 256 + 255) / 256, 256, 0, stream>>>(h1b, wf1,  bnf1, nullptr, h2b, B, 512,  256, 0);
    fc_kernel<<<(B * 64  + 255) / 256, 256, 0, stream>>>(h2b, wfin, nullptr, bfin, out, B, 256,  64,  1);
}